// GraphSAGE_27977416966302
// MI455X (gfx1250) — compile-verified
//
#include <hip/hip_runtime.h>

typedef __attribute__((ext_vector_type(2))) float v2f;
typedef __attribute__((ext_vector_type(8))) float v8f;

#define NNODES 50000
#define NEDGES 800000
#define KDIM   128

// ---------------------------------------------------------------- utilities
__global__ __launch_bounds__(256) void zero4_kernel(float4* __restrict__ p, int n4) {
  int i = blockIdx.x * blockDim.x + threadIdx.x;
  if (i < n4) p[i] = make_float4(0.f, 0.f, 0.f, 0.f);
}

__global__ __launch_bounds__(256) void degree_kernel(const int* __restrict__ dst,
                                                     float* __restrict__ deg) {
  int e = blockIdx.x * blockDim.x + threadIdx.x;
  if (e < NEDGES) atomicAdd(&deg[dst[e]], 1.0f);
}

__global__ __launch_bounds__(256) void invdeg_kernel(const float* __restrict__ deg,
                                                     float* __restrict__ inv) {
  int i = blockIdx.x * blockDim.x + threadIdx.x;
  if (i < NNODES) inv[i] = 1.0f / fmaxf(deg[i], 1.0f);
}

// Repack W [K x N] row-major -> WMMA B-fragment order: for each K-group of 4
// rows, the 4 values W[4g..4g+3][c] are contiguous:
//   P[((g*N + c) << 2) + (k & 3)] = W[k*N + c]
// This makes each B fragment a single aligned 8-byte LDS load.
__global__ __launch_bounds__(256) void repack_kernel(const float* __restrict__ W,
                                                     float* __restrict__ P, int N) {
  int i = blockIdx.x * blockDim.x + threadIdx.x;  // exactly K*N threads
  int k = i / N;
  int c = i % N;
  P[((((k >> 2) * N) + c) << 2) + (k & 3)] = W[i];
}

// scatter-add of projected features: agg[dst[e]] += T[src[e]] (CH channels)
template <int CH>
__global__ __launch_bounds__(256) void scatter_kernel(const float* __restrict__ T,
                                                      const int* __restrict__ src,
                                                      const int* __restrict__ dst,
                                                      float* __restrict__ agg) {
  constexpr int L = CH / 4;  // float4 slots per edge
  int t = blockIdx.x * blockDim.x + threadIdx.x;
  int e = t / L;
  int c = t % L;
  if (e >= NEDGES) return;
  int s = src[e];
  int d = dst[e];
  float4 v = ((const float4*)(T + s * CH))[c];
  float* o = agg + d * CH + c * 4;
  atomicAdd(o + 0, v.x);
  atomicAdd(o + 1, v.y);
  atomicAdd(o + 2, v.z);
  atomicAdd(o + 3, v.w);
}

// h = relu(S + inv_deg[node] * agg), in place into S   (128 channels)
__global__ __launch_bounds__(256) void fuse_relu_kernel(float* __restrict__ S,
                                                        const float* __restrict__ agg,
                                                        const float* __restrict__ inv) {
  int i = blockIdx.x * blockDim.x + threadIdx.x;  // exactly NNODES*128 threads
  float v = S[i] + inv[i >> 7] * agg[i];
  S[i] = fmaxf(v, 0.0f);
}

// out = S + inv_deg[node] * agg   (64 channels)
__global__ __launch_bounds__(256) void final_kernel(const float* __restrict__ S,
                                                    const float* __restrict__ agg,
                                                    const float* __restrict__ inv,
                                                    float* __restrict__ out) {
  int i = blockIdx.x * blockDim.x + threadIdx.x;  // exactly NNODES*64 threads
  out[i] = S[i] + inv[i >> 6] * agg[i];
}

// ------------------------------------------------------------------- GEMM
// Computes S = A @ Wself + bias  and  T = A @ Wneigh  in one pass over A.
// A: [50000 x 128] f32; Pself/Pneigh: fragment-packed weights (see repack).
// Block = 160 threads = 5 waves, one 16-row tile per wave; grid 625 blocks
// covers 3125 tiles exactly (no divergent guards -> EXEC all-ones for WMMA).
// Weights staged through LDS in two 64-row K-chunks (<=64 KB LDS); packed
// layout is chunk-contiguous, so staging stays float4-vectorized while each
// B fragment is one conflict-free ds_load_b64.
template <int N>
__global__ __launch_bounds__(160) void gemm_dual_kernel(
    const float* __restrict__ A,
    const float* __restrict__ Pself,
    const float* __restrict__ Pneigh,
    const float* __restrict__ bias,
    float* __restrict__ S,
    float* __restrict__ T) {
  constexpr int K  = KDIM;  // 128
  constexpr int KC = 64;    // K-chunk held in LDS
  constexpr int NT = N / 16;
  __shared__ float ldsW[2 * KC * N];  // [0,KC*N) = self chunk, rest = neigh

  const int tid   = threadIdx.x;
  const int lane  = tid & 31;
  const int wave  = tid >> 5;                       // 0..4
  const int l15   = lane & 15;
  const int khalf = (lane >> 4) * 2;                // 0 or 2 (A/B K split)
  const int rowBase = (blockIdx.x * 5 + wave) * 16; // < 50000 always

  // accumulators: NT self tiles (bias-seeded) + NT neigh tiles (zero)
  v8f acc[2 * NT];
#pragma unroll
  for (int ct = 0; ct < 2 * NT; ++ct) {
    float init = (ct < NT) ? bias[ct * 16 + l15] : 0.0f;
#pragma unroll
    for (int j = 0; j < 8; ++j) acc[ct][j] = init;
  }

  const float* Arow = A + (rowBase + l15) * K;

  for (int k0 = 0; k0 < K; k0 += KC) {
    __syncthreads();
    // packed chunk for rows [k0, k0+KC) is the contiguous KC*N-float region
    // starting at float offset k0*N
    for (int i = tid * 4; i < KC * N; i += 160 * 4) {
      *(float4*)&ldsW[i]          = *(const float4*)&Pself[k0 * N + i];
      *(float4*)&ldsW[KC * N + i] = *(const float4*)&Pneigh[k0 * N + i];
    }
    __syncthreads();

    for (int kk = 0; kk < KC; kk += 4) {
      // A fragment 16x4: lanes 0-15 -> K = kk,kk+1 ; lanes 16-31 -> kk+2,kk+3
      v2f a = *(const v2f*)&Arow[k0 + kk + khalf];
      const int g4 = (kk >> 2) * N;  // K-group base (in packed units of 4)
#pragma unroll
      for (int ct = 0; ct < 2 * NT; ++ct) {
        const int colB = (ct < NT ? ct : ct - NT) * 16 + l15;
        const float* Wb = &ldsW[(ct < NT ? 0 : KC * N) + ((g4 + colB) << 2) + khalf];
        v2f b = *(const v2f*)Wb;  // single ds_load_b64, even-aligned pair
        acc[ct] = __builtin_amdgcn_wmma_f32_16x16x4_f32(
            false, a, false, b, (short)0, acc[ct], false, false);
      }
    }
  }

  // C/D layout: VGPR j -> M = j + 8*(lane>>4), N = lane&15
#pragma unroll
  for (int ct = 0; ct < 2 * NT; ++ct) {
    float* O = (ct < NT ? S : T) + ((ct < NT ? ct : ct - NT) * 16 + l15);
    const int r0 = rowBase + (lane >> 4) * 8;
#pragma unroll
    for (int j = 0; j < 8; ++j) {
      O[(r0 + j) * N] = acc[ct][j];
    }
  }
}

// ----------------------------------------------------------------- launch
extern "C" void kernel_launch(void* const* d_in, const int* in_sizes, int n_in,
                              void* d_out, int out_size, void* d_ws, size_t ws_size,
                              hipStream_t stream) {
  (void)in_sizes; (void)n_in; (void)out_size; (void)ws_size;

  const float* x   = (const float*)d_in[0];
  const int*   src = (const int*)d_in[1];
  const int*   dst = (const int*)d_in[2];
  const float* Ws1 = (const float*)d_in[3];
  const float* Wn1 = (const float*)d_in[4];
  const float* b1  = (const float*)d_in[5];
  const float* Ws2 = (const float*)d_in[6];
  const float* Wn2 = (const float*)d_in[7];
  const float* b2  = (const float*)d_in[8];
  float* out = (float*)d_out;

  // workspace layout (floats); total ~19.35M floats = ~77.4 MB
  float* ws  = (float*)d_ws;
  float* deg = ws;                    // 50000
  float* inv = deg + NNODES;          // 50000
  float* S1  = inv + NNODES;          // 6.4M  (x@Wself1+b1, then h in place)
  float* T1  = S1 + NNODES * KDIM;    // 6.4M  (x@Wneigh1)
  float* agg = T1 + NNODES * KDIM;    // 6.4M  (agg1, reused as agg2)
  float* Ps1 = agg + NNODES * KDIM;   // 16384 packed weights
  float* Pn1 = Ps1 + KDIM * 128;      // 16384
  float* Ps2 = Pn1 + KDIM * 128;      // 8192
  float* Pn2 = Ps2 + KDIM * 64;       // 8192
  float* S2  = T1;                    // reuse T1: 3.2M (h@Wself2+b2)
  float* T2  = T1 + NNODES * 64;      // reuse T1: 3.2M (h@Wneigh2)

  // pack weights into WMMA B-fragment order (K*N threads, exact grids)
  repack_kernel<<<KDIM * 128 / 256, 256, 0, stream>>>(Ws1, Ps1, 128);
  repack_kernel<<<KDIM * 128 / 256, 256, 0, stream>>>(Wn1, Pn1, 128);
  repack_kernel<<<KDIM * 64 / 256, 256, 0, stream>>>(Ws2, Ps2, 64);
  repack_kernel<<<KDIM * 64 / 256, 256, 0, stream>>>(Wn2, Pn2, 64);

  // degrees -> inv_deg
  zero4_kernel<<<(NNODES / 4 + 255) / 256, 256, 0, stream>>>((float4*)deg, NNODES / 4);
  degree_kernel<<<NEDGES / 256, 256, 0, stream>>>(dst, deg);
  invdeg_kernel<<<(NNODES + 255) / 256, 256, 0, stream>>>(deg, inv);

  // layer 1: project first (linear agg commutes), then scatter-mean, relu
  gemm_dual_kernel<128><<<625, 160, 0, stream>>>(x, Ps1, Pn1, b1, S1, T1);
  zero4_kernel<<<NNODES * 128 / 4 / 256, 256, 0, stream>>>((float4*)agg, NNODES * 128 / 4);
  scatter_kernel<128><<<NEDGES * 32 / 256, 256, 0, stream>>>(T1, src, dst, agg);
  fuse_relu_kernel<<<NNODES * 128 / 256, 256, 0, stream>>>(S1, agg, inv);

  // layer 2 (64 output channels -> half the scatter traffic)
  gemm_dual_kernel<64><<<625, 160, 0, stream>>>(S1, Ps2, Pn2, b2, S2, T2);
  zero4_kernel<<<NNODES * 64 / 4 / 256, 256, 0, stream>>>((float4*)agg, NNODES * 64 / 4);
  scatter_kernel<64><<<NEDGES * 16 / 256, 256, 0, stream>>>(T2, src, dst, agg);
  final_kernel<<<NNODES * 64 / 256, 256, 0, stream>>>(S2, agg, inv, out);
}